// QSP_69767448756304
// MI455X (gfx1250) — compile-verified
//
#include <hip/hip_runtime.h>

typedef __attribute__((ext_vector_type(2))) float v2f;
typedef __attribute__((ext_vector_type(8))) float v8f;

#define NP  256     // Laurent exponent slots, exponent n = idx - 128
#define DEG 127

// ---------------------------------------------------------------------------
// Kernel 1: build Laurent coefficients of u00(theta) = sum_n a_n e^{i n theta}
// Recursion over row (a,b) of u:  a' = e^{+i phi}(p z + m z^-1),
//                                 b' = e^{-i phi}(p z - m z^-1),
// with p = (a+b)/2, m = (a-b)/2 (as Laurent polynomials).
// Then emit the WMMA A-operand coefficient table into d_ws:
//   Atab[t][g][n] = float2(cosCoef, sinCoef) for harmonic h = 4t+1+2g,
//   n = output row (0 = real, 1 = imag, 2..15 = zero padding).
// ---------------------------------------------------------------------------
__global__ __launch_bounds__(256) void qsp_coef_kernel(
    const float* __restrict__ phis, float* __restrict__ atab)
{
    __shared__ float aR[NP], aI[NP], bR[NP], bI[NP];
    __shared__ float pR[NP], pI[NP], mR[NP], mI[NP];
    const int n = threadIdx.x;

    float s0, c0;
    sincosf(phis[0], &s0, &c0);
    aR[n] = (n == 128) ? c0 : 0.f;      // a = e^{i phi0} at exponent 0
    aI[n] = (n == 128) ? s0 : 0.f;
    bR[n] = 0.f;
    bI[n] = 0.f;
    __syncthreads();

    for (int k = 1; k <= DEG; ++k) {
        float sp, cp;
        sincosf(phis[k], &sp, &cp);
        pR[n] = 0.5f * (aR[n] + bR[n]);
        pI[n] = 0.5f * (aI[n] + bI[n]);
        mR[n] = 0.5f * (aR[n] - bR[n]);
        mI[n] = 0.5f * (aI[n] - bI[n]);
        __syncthreads();
        const float pr = (n >= 1)      ? pR[n - 1] : 0.f;   // * z
        const float pi = (n >= 1)      ? pI[n - 1] : 0.f;
        const float mr = (n <= NP - 2) ? mR[n + 1] : 0.f;   // * z^-1
        const float mi = (n <= NP - 2) ? mI[n + 1] : 0.f;
        const float taR = pr + mr, taI = pi + mi;
        const float tbR = pr - mr, tbI = pi - mi;
        aR[n] =  taR * cp - taI * sp;      // * e^{+i phi}
        aI[n] =  taR * sp + taI * cp;
        bR[n] =  tbR * cp + tbI * sp;      // * e^{-i phi}
        bI[n] = -tbR * sp + tbI * cp;
        __syncthreads();
    }

    // Emit 1024 float2 entries: [t in 0..31][g in 0..1][n in 0..15]
    for (int j = 0; j < 4; ++j) {
        const int q   = n + 256 * j;
        const int t   = q >> 5;
        const int rem = q & 31;
        const int gg  = rem >> 4;
        const int col = rem & 15;
        const int h   = 4 * t + 1 + 2 * gg;       // odd harmonic, 1..127
        float cc = 0.f, sc = 0.f;
        if (col == 0) {            // real-output row:
            cc =  aR[128 + h] + aR[128 - h];      //  (ar_h + ar_-h) cos
            sc = -aI[128 + h] + aI[128 - h];      //  (ai_-h - ai_h) sin
        } else if (col == 1) {     // imag-output row:
            cc =  aI[128 + h] + aI[128 - h];      //  (ai_h + ai_-h) cos
            sc =  aR[128 + h] - aR[128 - h];      //  (ar_h - ar_-h) sin
        }
        atab[2 * q]     = cc;
        atab[2 * q + 1] = sc;
    }
}

// ---------------------------------------------------------------------------
// Kernel 2: per wave, evaluate 16 elements with 32 accumulating
// V_WMMA_F32_16X16X4_F32 ops:
//   A(16x4)  = coefficient chunk (rows: 0=real out, 1=imag out, rest zero)
//   B(4x16)  = basis values: lane l<16 -> {cos((4t+1)th_l), sin((4t+1)th_l)},
//              lane l+16   -> {cos((4t+3)th_l), sin((4t+3)th_l)}
//   basis advanced per chunk by a rotation of 4*theta (4 FMAs/lane).
// D: VGPR0 lanes 0-15 = real outputs, VGPR1 lanes 0-15 = imag outputs.
// ---------------------------------------------------------------------------
__global__ __launch_bounds__(256) void qsp_eval_kernel(
    const float* __restrict__ th, const float* __restrict__ atab,
    float* __restrict__ out, int B, int ngroups)
{
    __shared__ v2f Alds[1024];
    {
        const v2f* src = (const v2f*)atab;
        for (int i = threadIdx.x; i < 1024; i += 256) Alds[i] = src[i];
    }
    __syncthreads();

    const int lane   = threadIdx.x & 31;
    const int e      = lane & 15;        // element within group / A row M
    const int half   = lane >> 4;        // 0: harmonics 4t+1, 1: harmonics 4t+3
    const int wave   = (blockIdx.x * (blockDim.x >> 5)) + (threadIdx.x >> 5);
    const int nwaves = gridDim.x * (blockDim.x >> 5);
    const v2f* arow  = &Alds[half * 16 + e];

    for (int g = wave; g < ngroups; g += nwaves) {
        const int base = g << 4;
        const float theta = th[base + e];
        float s1, c1;
        sincosf(theta, &s1, &c1);
        const float c2 = c1 * c1 - s1 * s1;     // cos 2t
        const float s2 = 2.f * c1 * s1;         // sin 2t
        const float C4 = c2 * c2 - s2 * s2;     // chunk-step rotation: 4t
        const float S4 = 2.f * c2 * s2;
        float c = half ? (c1 * c2 - s1 * s2) : c1;   // cos((1+2*half) t)
        float s = half ? (s1 * c2 + c1 * s2) : s1;   // sin((1+2*half) t)

        v8f acc = {};
        #pragma unroll
        for (int t = 0; t < 32; ++t) {
            const v2f A = arow[t * 32];
            v2f Bv; Bv[0] = c; Bv[1] = s;
            acc = __builtin_amdgcn_wmma_f32_16x16x4_f32(
                false, A, false, Bv, (short)0, acc, false, false);
            const float cn = c * C4 - s * S4;
            const float sn = s * C4 + c * S4;
            c = cn; s = sn;
        }

        if (lane < 16) {
            out[base + lane]     = acc[0];   // D row M=0: real(u00)
            out[B + base + lane] = acc[1];   // D row M=1: imag(u00)
        }
    }
}

extern "C" void kernel_launch(void* const* d_in, const int* in_sizes, int n_in,
                              void* d_out, int out_size, void* d_ws, size_t ws_size,
                              hipStream_t stream) {
    const float* th   = (const float*)d_in[0];   // (B,) f32
    const float* phis = (const float*)d_in[1];   // (128,) f32
    float* out  = (float*)d_out;                 // [real(B) | imag(B)]
    float* atab = (float*)d_ws;                  // 8 KB coefficient table
    const int B = in_sizes[0];
    const int ngroups = B >> 4;

    qsp_coef_kernel<<<1, 256, 0, stream>>>(phis, atab);
    qsp_eval_kernel<<<2048, 256, 0, stream>>>(th, atab, out, B, ngroups);
}